// torch_kn_10050223473298
// MI455X (gfx1250) — compile-verified
//
#include <hip/hip_runtime.h>
#include <hip/hip_bf16.h>

typedef __attribute__((ext_vector_type(2))) float v2f;
typedef __attribute__((ext_vector_type(4))) float v4f;

__device__ __forceinline__ v2f splat2(float s) { v2f r = {s, s}; return r; }

// 7-term Horner on a float2 lane-pair, coeffs lowest-degree first.
// Vector*scalar ops splat; -ffp-contract=fast (HIP default) fuses to FMA,
// and the gfx1250 backend can pack pairs into v_pk_fma_f32.
__device__ __forceinline__ v2f poly7v(v2f t,
                                      float c0, float c1, float c2, float c3,
                                      float c4, float c5, float c6) {
    v2f r = splat2(c6);
    r = r * t + c5;
    r = r * t + c4;
    r = r * t + c3;
    r = r * t + c2;
    r = r * t + c1;
    r = r * t + c0;
    return r;
}

// Modified Bessel K1 on two lanes at once (A&S 9.8.7 / 9.8.8, fp32).
// TRANS budget: 1x v_log_f32 + 1x v_rcp_f32 + 1x v_exp_f32 per element:
//   - rcp(x) serves both ps/xs (valid where selected, since xs==x for x<=2)
//     and w = 2/x.
//   - exp(-x)/sqrt(x) = exp2(-x*log2e - 0.5*log2(x)) reuses the same log2.
__device__ __forceinline__ v2f k1v(v2f x) {
    const float LN2    = 0.69314718055994530942f;
    const float NLOG2E = -1.44269504088896340736f;

    v2f lg2, rx;
    lg2.x = __builtin_amdgcn_logf(x.x);   // v_log_f32 (log2)
    lg2.y = __builtin_amdgcn_logf(x.y);
    rx.x  = __builtin_amdgcn_rcpf(x.x);   // v_rcp_f32
    rx.y  = __builtin_amdgcn_rcpf(x.y);

    // ---- small-argument branch (x <= 2), safe arg xs = min(x, 2) ----
    v2f xs = x;
    xs.x = fminf(xs.x, 2.0f);
    xs.y = fminf(xs.y, 2.0f);
    v2f th = xs * 0.5f;
    v2f t  = th * th;                          // (xs/2)^2
    v2f tb = xs * 0.26666666666666666f;        // xs/3.75
    tb = tb * tb;                              // (xs/3.75)^2
    v2f i1 = xs * poly7v(tb, 0.5f, 0.87890594f, 0.51498869f, 0.15084934f,
                             0.02658733f, 0.00301532f, 0.00032411f);
    v2f ps = poly7v(t, 1.0f, 0.15443144f, -0.67278579f, -0.18156897f,
                       -0.01919402f, -0.00110404f, -0.00004686f);
    v2f lg = lg2 * LN2 - LN2;                  // ln(xs/2) when x<=2
    v2f small_ = lg * i1 + ps * rx;            // log*i1 + (1/xs)*poly

    // ---- large-argument branch (x > 2) ----
    v2f w  = rx * 2.0f;                        // 2/x
    v2f pl = poly7v(w, 1.25331414f, 0.23498619f, -0.03655620f, 0.01504268f,
                       -0.00780353f, 0.00325614f, -0.00068245f);
    v2f ea = x * NLOG2E - lg2 * 0.5f;          // -x*log2e - 0.5*log2(x)
    v2f e;
    e.x = __builtin_amdgcn_exp2f(ea.x);        // v_exp_f32
    e.y = __builtin_amdgcn_exp2f(ea.y);
    v2f large_ = e * pl;

    v2f res;
    res.x = (x.x <= 2.0f) ? small_.x : large_.x;
    res.y = (x.y <= 2.0f) ? small_.y : large_.y;
    return res;
}

#define ELEMS_PER_THREAD 8

__global__ __launch_bounds__(256) void bessel_k1_kernel(
        const float* __restrict__ in, float* __restrict__ out, long long n) {
    long long tid  = (long long)blockIdx.x * blockDim.x + threadIdx.x;
    long long base = tid * ELEMS_PER_THREAD;

    if (base + ELEMS_PER_THREAD <= n) {
        // Two 128-bit streaming loads: global_load_b128 th:TH_LOAD_NT
        const v4f* pin = (const v4f*)(in + base);
        v4f a = __builtin_nontemporal_load(pin);
        v4f b = __builtin_nontemporal_load(pin + 1);

        v2f r0 = k1v((v2f){a.x, a.y});
        v2f r1 = k1v((v2f){a.z, a.w});
        v2f r2 = k1v((v2f){b.x, b.y});
        v2f r3 = k1v((v2f){b.z, b.w});

        v4f ra = {r0.x, r0.y, r1.x, r1.y};
        v4f rb = {r2.x, r2.y, r3.x, r3.y};

        v4f* pout = (v4f*)(out + base);
        __builtin_nontemporal_store(ra, pout);
        __builtin_nontemporal_store(rb, pout + 1);
    } else if (base < n) {
        // Scalar tail (not taken for 8192*8192; kept for generality)
        for (long long i = base; i < n; ++i) {
            v2f xi = {in[i], in[i]};
            out[i] = k1v(xi).x;
        }
    }
}

extern "C" void kernel_launch(void* const* d_in, const int* in_sizes, int n_in,
                              void* d_out, int out_size, void* d_ws, size_t ws_size,
                              hipStream_t stream) {
    (void)n_in; (void)d_ws; (void)ws_size;
    const float* in = (const float*)d_in[0];
    float* out      = (float*)d_out;
    long long n     = (long long)out_size;   // 8192*8192 = 67,108,864

    const int block = 256;  // 8 wave32 waves per block
    long long threads = (n + ELEMS_PER_THREAD - 1) / ELEMS_PER_THREAD;
    int grid = (int)((threads + block - 1) / block);   // 32768 blocks

    bessel_k1_kernel<<<grid, block, 0, stream>>>(in, out, n);
}